// BVHSkeleton_62156766707902
// MI455X (gfx1250) — compile-verified
//
#include <hip/hip_runtime.h>
#include <stdint.h>

#define NJ    24
#define TPB   128          // bodies per block, 1 thread = 1 body (4 wave32s)
#define PADF  76           // padded floats per body in LDS (304 B, 16B-aligned, <=2-way bank conflict)
#define F4PB  18           // float4 chunks per body (72 floats = 24 joints * 3)

// SMPL kinematic tree (topologically ordered), and which joints have children.
__device__ constexpr int  kParent[NJ]   = {-1,0,0,0,1,2,3,4,5,6,7,8,9,9,9,12,13,14,16,17,18,19,20,21};
__device__ constexpr bool kHasChild[NJ] = { true,true,true,true,true,true,true,true,true,true,
                                            false,false,true,true,true,false,true,true,true,true,
                                            true,true,false,false };

__global__ __launch_bounds__(TPB) void fk_skeleton_kernel(
    const float* __restrict__ offsets,   // [24,3] uniform
    const float* __restrict__ poses,     // [B,24,3]
    const float* __restrict__ trans,     // [B,3]
    float* __restrict__ out,             // [B,24,3]
    int B)
{
    __shared__ __align__(16) float sBuf[TPB * PADF];

    const int tid  = threadIdx.x;
    const int b    = blockIdx.x * TPB + tid;
    const int nF4  = B * F4PB;
    const bool fullBlock = (blockIdx.x + 1) * (TPB * F4PB) <= nF4;  // uniform
    const unsigned sBase = (unsigned)(uintptr_t)&sBuf[0];           // LDS byte offset
    const unsigned long long posesS = (unsigned long long)(uintptr_t)poses;
    const unsigned long long outS   = (unsigned long long)(uintptr_t)out;

    // ---- Stage poses: global -> LDS, coalesced async b128 copies (ASYNCcnt path) ----
    if (fullBlock) {
#pragma unroll
        for (int it = 0; it < F4PB; ++it) {
            const int idx  = it * TPB + tid;                   // float4 index within block
            const int gidx = blockIdx.x * (TPB * F4PB) + idx;  // float4 index within tensor
            const int body = idx / F4PB;
            const int q    = idx - body * F4PB;
            const unsigned lds  = sBase + (unsigned)(body * (PADF * 4) + q * 16);
            const unsigned voff = (unsigned)gidx * 16u;
            asm volatile("global_load_async_to_lds_b128 %0, %1, %2"
                         :: "v"(lds), "v"(voff), "s"(posesS) : "memory");
        }
    } else {
#pragma unroll
        for (int it = 0; it < F4PB; ++it) {
            const int idx  = it * TPB + tid;
            const int gidx = blockIdx.x * (TPB * F4PB) + idx;
            if (gidx < nF4) {
                const int body = idx / F4PB;
                const int q    = idx - body * F4PB;
                const unsigned lds  = sBase + (unsigned)(body * (PADF * 4) + q * 16);
                const unsigned voff = (unsigned)gidx * 16u;
                asm volatile("global_load_async_to_lds_b128 %0, %1, %2"
                             :: "v"(lds), "v"(voff), "s"(posesS) : "memory");
            }
        }
    }

    // Overlap: issue the per-body translation loads while the async DMA runs.
    const int   bc  = (b < B) ? b : (B - 1);
    const float trx = trans[bc * 3 + 0];
    const float trY = trans[bc * 3 + 1];
    const float trz = trans[bc * 3 + 2];

    asm volatile("s_wait_asynccnt 0" ::: "memory");
    __syncthreads();

    // ---- Per-body forward kinematics (fully unrolled; tree indices are constants) ----
    const int lb = tid * PADF;

    float Rg[NJ][9];
    float tg[NJ][3];

#pragma unroll
    for (int j = 0; j < NJ; ++j) {
        const int   p  = kParent[j];
        const float ox = offsets[j * 3 + 0];
        const float oy = offsets[j * 3 + 1];
        const float oz = offsets[j * 3 + 2];

        if (j == 0) {
            tg[0][0] = ox; tg[0][1] = oy; tg[0][2] = oz;
        } else {
            tg[j][0] = __fmaf_rn(Rg[p][0], ox, __fmaf_rn(Rg[p][1], oy, __fmaf_rn(Rg[p][2], oz, tg[p][0])));
            tg[j][1] = __fmaf_rn(Rg[p][3], ox, __fmaf_rn(Rg[p][4], oy, __fmaf_rn(Rg[p][5], oz, tg[p][1])));
            tg[j][2] = __fmaf_rn(Rg[p][6], ox, __fmaf_rn(Rg[p][7], oy, __fmaf_rn(Rg[p][8], oz, tg[p][2])));
        }

        if (kHasChild[j]) {
            // Rodrigues (matches reference: angle = |v + 1e-8|, dir = v / angle)
            const float vx = sBuf[lb + j * 3 + 0];
            const float vy = sBuf[lb + j * 3 + 1];
            const float vz = sBuf[lb + j * 3 + 2];
            const float axv = vx + 1e-8f, ayv = vy + 1e-8f, azv = vz + 1e-8f;
            const float ang = __builtin_amdgcn_sqrtf(axv * axv + ayv * ayv + azv * azv);
            const float inv = __builtin_amdgcn_rcpf(ang);
            float s, c;
            __sincosf(ang, &s, &c);
            const float dx = vx * inv, dy = vy * inv, dz = vz * inv;
            const float qn = dx * dx + dy * dy + dz * dz;
            const float t1 = 1.0f - c;
            const float r0 = 1.0f + t1 * (dx * dx - qn);
            const float r1 = t1 * dx * dy - s * dz;
            const float r2 = t1 * dx * dz + s * dy;
            const float r3 = t1 * dx * dy + s * dz;
            const float r4 = 1.0f + t1 * (dy * dy - qn);
            const float r5 = t1 * dy * dz - s * dx;
            const float r6 = t1 * dx * dz - s * dy;
            const float r7 = t1 * dy * dz + s * dx;
            const float r8 = 1.0f + t1 * (dz * dz - qn);

            if (j == 0) {
                Rg[0][0] = r0; Rg[0][1] = r1; Rg[0][2] = r2;
                Rg[0][3] = r3; Rg[0][4] = r4; Rg[0][5] = r5;
                Rg[0][6] = r6; Rg[0][7] = r7; Rg[0][8] = r8;
            } else {
                Rg[j][0] = Rg[p][0] * r0 + Rg[p][1] * r3 + Rg[p][2] * r6;
                Rg[j][1] = Rg[p][0] * r1 + Rg[p][1] * r4 + Rg[p][2] * r7;
                Rg[j][2] = Rg[p][0] * r2 + Rg[p][1] * r5 + Rg[p][2] * r8;
                Rg[j][3] = Rg[p][3] * r0 + Rg[p][4] * r3 + Rg[p][5] * r6;
                Rg[j][4] = Rg[p][3] * r1 + Rg[p][4] * r4 + Rg[p][5] * r7;
                Rg[j][5] = Rg[p][3] * r2 + Rg[p][4] * r5 + Rg[p][5] * r8;
                Rg[j][6] = Rg[p][6] * r0 + Rg[p][7] * r3 + Rg[p][8] * r6;
                Rg[j][7] = Rg[p][6] * r1 + Rg[p][7] * r4 + Rg[p][8] * r7;
                Rg[j][8] = Rg[p][6] * r2 + Rg[p][7] * r5 + Rg[p][8] * r8;
            }
        }

        // Pose slot j is fully consumed -> reuse it in place for the output position.
        sBuf[lb + j * 3 + 0] = tg[j][0] + trx;
        sBuf[lb + j * 3 + 1] = tg[j][1] + trY;
        sBuf[lb + j * 3 + 2] = tg[j][2] + trz;
    }

    __syncthreads();

    // ---- Store positions: LDS -> global, coalesced async b128 copies ----
    if (fullBlock) {
#pragma unroll
        for (int it = 0; it < F4PB; ++it) {
            const int idx  = it * TPB + tid;
            const int gidx = blockIdx.x * (TPB * F4PB) + idx;
            const int body = idx / F4PB;
            const int q    = idx - body * F4PB;
            const unsigned lds  = sBase + (unsigned)(body * (PADF * 4) + q * 16);
            const unsigned voff = (unsigned)gidx * 16u;
            asm volatile("global_store_async_from_lds_b128 %0, %1, %2"
                         :: "v"(voff), "v"(lds), "s"(outS) : "memory");
        }
    } else {
#pragma unroll
        for (int it = 0; it < F4PB; ++it) {
            const int idx  = it * TPB + tid;
            const int gidx = blockIdx.x * (TPB * F4PB) + idx;
            if (gidx < nF4) {
                const int body = idx / F4PB;
                const int q    = idx - body * F4PB;
                const unsigned lds  = sBase + (unsigned)(body * (PADF * 4) + q * 16);
                const unsigned voff = (unsigned)gidx * 16u;
                asm volatile("global_store_async_from_lds_b128 %0, %1, %2"
                             :: "v"(voff), "v"(lds), "s"(outS) : "memory");
            }
        }
    }
    // S_ENDPGM performs an implicit wait-idle, but drain ASYNCcnt explicitly for clarity.
    asm volatile("s_wait_asynccnt 0" ::: "memory");
}

extern "C" void kernel_launch(void* const* d_in, const int* in_sizes, int n_in,
                              void* d_out, int out_size, void* d_ws, size_t ws_size,
                              hipStream_t stream) {
    const float* offsets = (const float*)d_in[0];   // [24,3]
    const float* poses   = (const float*)d_in[1];   // [B,24,3]
    const float* trans   = (const float*)d_in[2];   // [B,3]
    float* out = (float*)d_out;                     // [B,24,3]

    const int B    = in_sizes[1] / (NJ * 3);
    const int grid = (B + TPB - 1) / TPB;
    fk_skeleton_kernel<<<dim3(grid), dim3(TPB), 0, stream>>>(offsets, poses, trans, out, B);
}